// TrellisQuantizer_61057255080571
// MI455X (gfx1250) — compile-verified
//
#include <hip/hip_runtime.h>

typedef __attribute__((ext_vector_type(2))) float v2f;
typedef __attribute__((ext_vector_type(8))) float v8f;

// Problem constants (from reference: L=16, K=2, V=2)
#define S_ 65536   // number of states
#define R_ 4096    // reduced states (S >> 4)
#define D_ 16      // candidates per reduced state
#define B_ 512     // batch
#define LOG2R_ 12

// ---------------------------------------------------------------------------
// Kernel 1: best_val[b,r] = min_d cost[b, sc[r,d]];  prev_state[b,r] = winner.
// Pure streaming pass over cost (128 MB read once). Consecutive lanes map to
// consecutive r, so each of the 16 gathered cost loads is fully coalesced
// (candidates are contiguous in r at stride 4096). sc rows are read via int4
// (16B) loads; the 256 KB table lives in L2.
// ---------------------------------------------------------------------------
__global__ __launch_bounds__(256) void trellis_min_kernel(
    const float* __restrict__ cost,        // (B, S)
    const int*   __restrict__ sc,          // (R, D)
    float*       __restrict__ best_val,    // (B, R)  [workspace]
    int*         __restrict__ prev_state)  // (B, R)  [output 0]
{
    const int t = blockIdx.x * blockDim.x + threadIdx.x;  // [0, B*R)
    const int r = t & (R_ - 1);
    const int b = t >> LOG2R_;

    const int4*  sc4  = (const int4*)(sc + r * D_);
    const float* crow = cost + (size_t)b * S_;

    float bv = 3.402823466e+38f;
    int   bs = 0;

#pragma unroll
    for (int q = 0; q < 4; ++q) {
        const int4 idx = sc4[q];
        float v;
        v = crow[idx.x]; if (v < bv) { bv = v; bs = idx.x; }
        v = crow[idx.y]; if (v < bv) { bv = v; bs = idx.y; }
        v = crow[idx.z]; if (v < bv) { bv = v; bs = idx.z; }
        v = crow[idx.w]; if (v < bv) { bv = v; bs = idx.w; }
    }

    best_val[t]   = bv;
    prev_state[t] = bs;
}

// ---------------------------------------------------------------------------
// Kernel 2: new_cost[b, s] = ||lut[s] - orig[b]||^2 + best_val[b, s>>4]
// via V_WMMA_F32_16X16X4_F32:
//   A (16x4, M=batch):  [ -2*ox, -2*oy, ox^2+oy^2, 1 ]
//   B (4x16, N=state):  [ lx, ly, 1, lx^2+ly^2 ]^T
//   C (16x16):          best_val[b0+m, r] broadcast along N (tile = one r)
//   D = A*B + C  ==  new_cost tile, stored lane-contiguous along s.
// Operand packing per ISA 32-bit layouts:
//   A: lanes 0-15 -> M=lane, K={0,1}; lanes 16-31 -> M=lane-16, K={2,3}
//   B: lanes 0-15 -> N=lane, K={0,1}; lanes 16-31 -> N=lane-16, K={2,3}
//   C/D: vgpr v, lanes 0-15 -> (M=v, N=lane); lanes 16-31 -> (M=v+8, N=lane-16)
// Each wave fixes a 16-row batch tile (A is loop-invariant) and sweeps 8
// consecutive 16-state tiles.
// ---------------------------------------------------------------------------
#define TPW 8  // s-tiles per wave

__global__ __launch_bounds__(256) void trellis_newcost_wmma_kernel(
    const float* __restrict__ lut,        // (S, 2)
    const float* __restrict__ orig,       // (B, 2)
    const float* __restrict__ best_val,   // (B, R)
    float*       __restrict__ new_cost)   // (B, S)  [output 1]
{
    const int lane = threadIdx.x & 31;
    const int wave = threadIdx.x >> 5;        // 0..7
    const int half = lane >> 4;               // 0: K={0,1}, 1: K={2,3}
    const int ln   = lane & 15;

    const int b0    = blockIdx.y << 4;                  // batch tile base
    const int tile0 = (blockIdx.x * 8 + wave) * TPW;    // first r of this wave

    // ---- A operand (loop-invariant) ----
    const float ox = orig[(b0 + ln) * 2 + 0];
    const float oy = orig[(b0 + ln) * 2 + 1];
    v2f a;
    a.x = (half == 0) ? (-2.0f * ox) : (ox * ox + oy * oy);
    a.y = (half == 0) ? (-2.0f * oy) : 1.0f;

    const int mrow = b0 + half * 8;  // first batch row this lane-half owns in C/D

    for (int tt = 0; tt < TPW; ++tt) {
        const int r  = tile0 + tt;
        const int s0 = r << 4;

        // ---- B operand ----
        const float lx = lut[(s0 + ln) * 2 + 0];
        const float ly = lut[(s0 + ln) * 2 + 1];
        v2f bm;
        bm.x = (half == 0) ? lx : 1.0f;
        bm.y = (half == 0) ? ly : (lx * lx + ly * ly);

        // ---- C operand: best_val[b0+m, r], broadcast along N ----
        const float* bvp = best_val + (size_t)mrow * R_ + r;
        v8f c;
#pragma unroll
        for (int v = 0; v < 8; ++v) c[v] = bvp[(size_t)v * R_];

        // D = A*B + C   (emits v_wmma_f32_16x16x4_f32)
        v8f d = __builtin_amdgcn_wmma_f32_16x16x4_f32(
            /*neg_a=*/false, a, /*neg_b=*/false, bm,
            /*c_mod=*/(short)0, c, /*reuse_a=*/false, /*reuse_b=*/false);

        // ---- store: row m = mrow+v, cols s0..s0+15 (lane-contiguous) ----
        float* out = new_cost + (size_t)mrow * S_ + s0 + ln;
#pragma unroll
        for (int v = 0; v < 8; ++v) out[(size_t)v * S_] = d[v];
    }
}

// ---------------------------------------------------------------------------
// Launch. d_in order: training_lut (S,2) f32, cost (B,S) f32,
// orig_seq_part (B,2) f32, state_candidates (R,D) i32.
// d_out: prev_state (B*R int32) then new_cost (B*S f32), concatenated flat.
// d_ws: best_val (B*R f32 = 8 MB).
// ---------------------------------------------------------------------------
extern "C" void kernel_launch(void* const* d_in, const int* in_sizes, int n_in,
                              void* d_out, int out_size, void* d_ws, size_t ws_size,
                              hipStream_t stream) {
    const float* lut  = (const float*)d_in[0];
    const float* cost = (const float*)d_in[1];
    const float* orig = (const float*)d_in[2];
    const int*   sc   = (const int*)d_in[3];

    int*   prev_state = (int*)d_out;
    float* new_cost   = (float*)d_out + (size_t)B_ * R_;
    float* best_val   = (float*)d_ws;

    // Pass 1: 2M threads, one (b,r) each.
    trellis_min_kernel<<<dim3((B_ * R_) / 256), dim3(256), 0, stream>>>(
        cost, sc, best_val, prev_state);

    // Pass 2: grid = (R/(8 waves * TPW), B/16) = (64, 32), 8 waves/block.
    trellis_newcost_wmma_kernel<<<dim3(R_ / (8 * TPW), B_ / 16), dim3(256), 0, stream>>>(
        lut, orig, best_val, new_cost);
}